// RWKV_41755672052470
// MI455X (gfx1250) — compile-verified
//
#include <hip/hip_runtime.h>

typedef __attribute__((ext_vector_type(2))) float v2f;
typedef __attribute__((ext_vector_type(8))) float v8f;

#define B_     16
#define T_     1024
#define CIN    64
#define DATTN  256
#define NEMBD  128
#define NH     4
#define HS     64
#define DEPTH_ 8

__device__ __forceinline__ v8f wmma4(v2f a, v2f b, v8f c) {
  // D = A(16x4 f32) * B(4x16 f32) + C(16x16 f32)
  return __builtin_amdgcn_wmma_f32_16x16x4_f32(false, a, false, b, (short)0, c, false, false);
}

// ---------------------------------------------------------------------------
// qkv = x[M=B*T,64] @ w_qkv[64,768] + b_qkv ; scatter into q/k/v [B,H,T,HS]
// one 16x16 C tile per wave, 8 waves per block
// ---------------------------------------------------------------------------
__global__ void qkv_gemm(const float* __restrict__ x, const float* __restrict__ w,
                         const float* __restrict__ bias,
                         float* __restrict__ q, float* __restrict__ k, float* __restrict__ v) {
  const int lane  = threadIdx.x & 31;
  const int wave  = threadIdx.x >> 5;
  const int tile  = blockIdx.x * 8 + wave;
  const int TN    = 768 / 16;                // 48 col tiles
  const int m0    = (tile / TN) * 16;
  const int n0    = (tile % TN) * 16;
  const int mlane = lane & 15;
  const int koff  = (lane >> 4) << 1;        // 0 or 2

  v8f acc = {0,0,0,0,0,0,0,0};
  const float* arow = x + (size_t)(m0 + mlane) * CIN;
  for (int kk = 0; kk < CIN; kk += 4) {
    v2f a, b;
    a.x = arow[kk + koff];
    a.y = arow[kk + koff + 1];
    b.x = w[(size_t)(kk + koff)     * 768 + n0 + mlane];
    b.y = w[(size_t)(kk + koff + 1) * 768 + n0 + mlane];
    acc = wmma4(a, b, acc);
  }

  const int col = n0 + mlane;
  const int seg = col >> 8;                  // 0=q 1=k 2=v
  const int h   = (col >> 6) & 3;
  const int d   = col & 63;
  float* dst = (seg == 0) ? q : (seg == 1) ? k : v;
  const float bb = bias[col];
  const int rowoff = (lane >> 4) << 3;       // lanes 16-31 hold rows M=r+8
  for (int r = 0; r < 8; ++r) {
    const int row = m0 + r + rowoff;
    const int b_  = row >> 10, t = row & 1023;
    dst[(((size_t)(b_ * NH + h)) * T_ + t) * HS + d] = acc[r] + bb;
  }
}

// ---------------------------------------------------------------------------
// Flash attention: one (b,h, q-tile of 16 rows) per wave, online softmax.
// q,k,v in [B*H, T, HS]; o written as [B*T, 256] row-major.
// ---------------------------------------------------------------------------
__global__ void flash_attn(const float* __restrict__ q, const float* __restrict__ k,
                           const float* __restrict__ v, float* __restrict__ o) {
  __shared__ float pbuf[8][16][17];          // per-wave P tile (C-layout -> A-layout bounce)
  const int lane  = threadIdx.x & 31;
  const int wave  = threadIdx.x >> 5;
  const int unit  = blockIdx.x * 8 + wave;
  const int bh    = unit >> 6;               // T_/16 = 64 q-tiles
  const int qt    = unit & 63;
  const int t0    = qt * 16;
  const float* qb = q + (size_t)bh * T_ * HS;
  const float* kb = k + (size_t)bh * T_ * HS;
  const float* vb = v + (size_t)bh * T_ * HS;
  const int mlane  = lane & 15;
  const int koff   = (lane >> 4) << 1;
  const int rowoff = (lane >> 4) << 3;

  // Preload Q A-fragments for all 16 K-steps (HS=64 / 4)
  v2f qa[16];
  {
    const float* qrow = qb + (size_t)(t0 + mlane) * HS;
    for (int i = 0; i < 16; ++i) { qa[i].x = qrow[4*i + koff]; qa[i].y = qrow[4*i + koff + 1]; }
  }

  const v8f vzero = {0,0,0,0,0,0,0,0};
  v8f oacc[4] = {vzero, vzero, vzero, vzero};
  float mrow[8], lrow[8];
  for (int r = 0; r < 8; ++r) { mrow[r] = -3.0e38f; lrow[r] = 0.0f; }

  for (int kt = 0; kt < T_; kt += 16) {
    // S = Q @ K^T   (B operand of K^T == K rows, transpose-symmetric layout)
    v8f s = vzero;
    const float* krow = kb + (size_t)(kt + mlane) * HS;
    for (int i = 0; i < 16; ++i) {
      v2f b; b.x = krow[4*i + koff]; b.y = krow[4*i + koff + 1];
      s = wmma4(qa[i], b, s);
    }
    // online softmax; each C row lives across the 16 lanes of one half
    for (int r = 0; r < 8; ++r) {
      float sv = s[r] * 0.125f;              // 1/sqrt(64)
      float mx = sv;
      mx = fmaxf(mx, __shfl_xor(mx, 1, 32));
      mx = fmaxf(mx, __shfl_xor(mx, 2, 32));
      mx = fmaxf(mx, __shfl_xor(mx, 4, 32));
      mx = fmaxf(mx, __shfl_xor(mx, 8, 32));
      const float mnew  = fmaxf(mrow[r], mx);
      const float alpha = __expf(mrow[r] - mnew);
      const float p     = __expf(sv - mnew);
      float ps = p;
      ps += __shfl_xor(ps, 1, 32);
      ps += __shfl_xor(ps, 2, 32);
      ps += __shfl_xor(ps, 4, 32);
      ps += __shfl_xor(ps, 8, 32);
      lrow[r] = lrow[r] * alpha + ps;
      mrow[r] = mnew;
      for (int tc = 0; tc < 4; ++tc) oacc[tc][r] *= alpha;
      pbuf[wave][r + rowoff][mlane] = p;     // C-layout store
    }
    // O += P(16x16) @ V(16x64); P re-read in A-layout from LDS (DScnt orders it)
    for (int kk = 0; kk < 16; kk += 4) {
      v2f pa;
      pa.x = pbuf[wave][mlane][kk + koff];
      pa.y = pbuf[wave][mlane][kk + koff + 1];
      for (int tc = 0; tc < 4; ++tc) {
        v2f b;
        b.x = vb[(size_t)(kt + kk + koff)     * HS + tc*16 + mlane];
        b.y = vb[(size_t)(kt + kk + koff + 1) * HS + tc*16 + mlane];
        oacc[tc] = wmma4(pa, b, oacc[tc]);
      }
    }
  }
  // epilogue: normalize, write o[B*T,256]
  const int b_ = bh >> 2, h = bh & 3;
  for (int r = 0; r < 8; ++r) {
    const float inv = 1.0f / lrow[r];
    const int t = t0 + r + rowoff;
    float* orow = o + ((size_t)(b_ * T_ + t)) * DATTN + h * HS;
    for (int tc = 0; tc < 4; ++tc) orow[tc*16 + mlane] = oacc[tc][r] * inv;
  }
}

// ---------------------------------------------------------------------------
// x_next = o[M=B*T,256] @ w_fused[256,64] + b_fused
// ---------------------------------------------------------------------------
__global__ void proj_gemm(const float* __restrict__ o, const float* __restrict__ wf,
                          const float* __restrict__ bf, float* __restrict__ xout) {
  const int lane  = threadIdx.x & 31;
  const int wave  = threadIdx.x >> 5;
  const int tile  = blockIdx.x * 8 + wave;
  const int TN    = CIN / 16;                // 4
  const int m0    = (tile / TN) * 16;
  const int n0    = (tile % TN) * 16;
  const int mlane = lane & 15;
  const int koff  = (lane >> 4) << 1;

  v8f acc = {0,0,0,0,0,0,0,0};
  const float* arow = o + (size_t)(m0 + mlane) * DATTN;
  for (int kk = 0; kk < DATTN; kk += 4) {
    v2f a, b;
    a.x = arow[kk + koff];
    a.y = arow[kk + koff + 1];
    b.x = wf[(size_t)(kk + koff)     * CIN + n0 + mlane];
    b.y = wf[(size_t)(kk + koff + 1) * CIN + n0 + mlane];
    acc = wmma4(a, b, acc);
  }
  const int col = n0 + mlane;
  const float bb = bf[col];
  const int rowoff = (lane >> 4) << 3;
  for (int r = 0; r < 8; ++r)
    xout[(size_t)(m0 + r + rowoff) * CIN + col] = acc[r] + bb;
}

// w_fused = w_out @ w_adj ; b_fused = b_out @ w_adj + b_adj  (tiny, once)
__global__ void fuse_weights(const float* __restrict__ w_out, const float* __restrict__ b_out,
                             const float* __restrict__ w_adj, const float* __restrict__ b_adj,
                             float* __restrict__ wf, float* __restrict__ bf) {
  const int i = threadIdx.x;                 // 0..255 rows of w_out
  for (int n = 0; n < CIN; ++n) {
    float acc = 0.0f;
    for (int j = 0; j < NEMBD; ++j) acc += w_out[i * NEMBD + j] * w_adj[j * CIN + n];
    wf[i * CIN + n] = acc;
  }
  if (i < CIN) {
    float acc = b_adj[i];
    for (int j = 0; j < NEMBD; ++j) acc += b_out[j] * w_adj[j * CIN + i];
    bf[i] = acc;
  }
}

__global__ void copy_center(const float* __restrict__ xin, float* __restrict__ c1,
                            float* __restrict__ c2, int d) {
  const int i = threadIdx.x;                 // 1024 = B_*CIN
  const int b = i >> 6, ch = i & 63;
  const float vv = xin[((size_t)b * T_ + T_/2) * CIN + ch];
  c1[((size_t)b * DEPTH_ + d) * CIN + ch] = vv;
  c2[((size_t)b * DEPTH_ + d) * CIN + ch] = vv;
}

extern "C" void kernel_launch(void* const* d_in, const int* in_sizes, int n_in,
                              void* d_out, int out_size, void* d_ws, size_t ws_size,
                              hipStream_t stream) {
  const float* x_in  = (const float*)d_in[0];
  const float* w_qkv = (const float*)d_in[1];
  const float* b_qkv = (const float*)d_in[2];
  const float* w_out = (const float*)d_in[3];
  const float* b_out = (const float*)d_in[4];
  const float* w_adj = (const float*)d_in[5];
  const float* b_adj = (const float*)d_in[6];

  float* out_x  = (float*)d_out;                          // [16,1024,64]
  float* out_c1 = out_x  + (size_t)B_ * T_ * CIN;         // [16,8,64]
  float* out_c2 = out_c1 + (size_t)B_ * DEPTH_ * CIN;     // [16,8,64]

  float* ws = (float*)d_ws;
  size_t off = 0;
  float* xbuf0 = ws + off; off += (size_t)B_ * T_ * CIN;
  float* xbuf1 = ws + off; off += (size_t)B_ * T_ * CIN;
  float* qb    = ws + off; off += (size_t)B_ * NH * T_ * HS;
  float* kb    = ws + off; off += (size_t)B_ * NH * T_ * HS;
  float* vb    = ws + off; off += (size_t)B_ * NH * T_ * HS;
  float* ob    = ws + off; off += (size_t)B_ * T_ * DATTN;
  float* wf    = ws + off; off += (size_t)DATTN * CIN;
  float* bf    = ws + off; off += CIN;

  fuse_weights<<<1, 256, 0, stream>>>(w_out, b_out, w_adj, b_adj, wf, bf);

  float* xb[2] = {xbuf0, xbuf1};
  for (int d = 0; d < DEPTH_; ++d) {
    const float* xin = (d == 0) ? x_in : xb[d & 1];
    float* xout = (d == DEPTH_ - 1) ? out_x : xb[(d + 1) & 1];

    copy_center<<<1, B_ * CIN, 0, stream>>>(xin, out_c1, out_c2, d);
    // (B*T/16)*(768/16) = 49152 tiles / 8 waves = 6144 blocks
    qkv_gemm<<<(B_ * T_ / 16) * (768 / 16) / 8, 256, 0, stream>>>(xin, w_qkv, b_qkv, qb, kb, vb);
    // 64 bh * 64 q-tiles = 4096 wave-units / 8 = 512 blocks
    flash_attn<<<(B_ * NH) * (T_ / 16) / 8, 256, 0, stream>>>(qb, kb, vb, ob);
    // 1024*4 tiles / 8 = 512 blocks
    proj_gemm<<<(B_ * T_ / 16) * (CIN / 16) / 8, 256, 0, stream>>>(ob, wf, bf, xout);
  }
}